// pnt2_cls_ssg_38611755991296
// MI455X (gfx1250) — compile-verified
//
#include <hip/hip_runtime.h>
#include <hip/hip_bf16.h>
#include <stddef.h>

typedef __attribute__((ext_vector_type(2))) float v2f;
typedef __attribute__((ext_vector_type(8))) float v8f;

#define BN_EPS 1e-5f

__device__ __forceinline__ v8f wmma4(v2f a, v2f b, v8f c) {
  // V_WMMA_F32_16X16X4_F32 : D = A(16x4) * B(4x16) + C(16x16), all fp32
  return __builtin_amdgcn_wmma_f32_16x16x4_f32(false, a, false, b, (short)0, c, false, false);
}

// ---------------- BN fold: s = g/sqrt(v+eps), t = b - m*s ----------------
__global__ void bn_fold_kernel(const float* __restrict__ g, const float* __restrict__ b,
                               const float* __restrict__ m, const float* __restrict__ v,
                               float* __restrict__ s, float* __restrict__ t, int C) {
  int i = blockIdx.x * blockDim.x + threadIdx.x;
  if (i < C) {
    float sc = g[i] / sqrtf(v[i] + BN_EPS);
    s[i] = sc;
    t[i] = b[i] - m[i] * sc;
  }
}

// ---------------- Farthest point sampling (one block per batch) ----------------
__global__ __launch_bounds__(256) void fps_kernel(const float* __restrict__ src, int stride,
                                                  int N, int npoint, int* __restrict__ idx_out) {
  __shared__ float rv[256];
  __shared__ int   ri[256];
  const int b = blockIdx.x, tid = threadIdx.x;
  const float* xb = src + (size_t)b * N * stride;
  float dloc[16];
  #pragma unroll
  for (int j = 0; j < 16; ++j) dloc[j] = 1e10f;
  int last = 0;
  if (tid == 0) idx_out[(size_t)b * npoint] = 0;
  for (int it = 1; it < npoint; ++it) {
    float lx = xb[(size_t)last * stride + 0];
    float ly = xb[(size_t)last * stride + 1];
    float lz = xb[(size_t)last * stride + 2];
    float bestv = -1.0f; int besti = 0;
    for (int j = 0; j < 16; ++j) {
      int n = tid + j * 256;
      if (n >= N) break;
      float dx = xb[(size_t)n * stride + 0] - lx;
      float dy = xb[(size_t)n * stride + 1] - ly;
      float dz = xb[(size_t)n * stride + 2] - lz;
      float d = dx * dx + dy * dy + dz * dz;
      float dd = fminf(dloc[j], d);
      dloc[j] = dd;
      if (dd > bestv) { bestv = dd; besti = n; }   // ascending n -> first-max tie rule
    }
    rv[tid] = bestv; ri[tid] = besti;
    __syncthreads();
    for (int off = 128; off > 0; off >>= 1) {
      if (tid < off) {
        float v2 = rv[tid + off]; int i2 = ri[tid + off];
        if (v2 > rv[tid] || (v2 == rv[tid] && i2 < ri[tid])) { rv[tid] = v2; ri[tid] = i2; }
      }
      __syncthreads();
    }
    last = ri[0];
    if (tid == 0) idx_out[(size_t)b * npoint + it] = last;
    __syncthreads();
  }
}

// ---------------- gather selected centroids ----------------
__global__ void gather_xyz_kernel(const float* __restrict__ src, int stride, int N,
                                  const int* __restrict__ idx, int S, float* __restrict__ out) {
  int i = blockIdx.x * blockDim.x + threadIdx.x;
  if (i >= 16 * S) return;
  int b = i / S, s = i % S;
  int pi = idx[(size_t)b * S + s];
  const float* p = src + ((size_t)b * N + pi) * stride;
  float* o = out + ((size_t)b * S + s) * 3;
  o[0] = p[0]; o[1] = p[1]; o[2] = p[2];
}

// ---------------- ball query: first ns hits in ascending order, pad with first ----------------
__global__ void ball_query_kernel(const float* __restrict__ src, int stride, int N,
                                  const float* __restrict__ q, int S, float r2, int ns,
                                  int* __restrict__ out) {
  int i = blockIdx.x * blockDim.x + threadIdx.x;
  if (i >= 16 * S) return;
  int b = i / S, s = i % S;
  float qx = q[((size_t)b * S + s) * 3 + 0];
  float qy = q[((size_t)b * S + s) * 3 + 1];
  float qz = q[((size_t)b * S + s) * 3 + 2];
  const float* xb = src + (size_t)b * N * stride;
  int* ob = out + ((size_t)b * S + s) * ns;
  int cnt = 0, first = 0;
  for (int n = 0; n < N && cnt < ns; ++n) {
    float dx = xb[(size_t)n * stride + 0] - qx;
    float dy = xb[(size_t)n * stride + 1] - qy;
    float dz = xb[(size_t)n * stride + 2] - qz;
    if (dx * dx + dy * dy + dz * dz < r2) {
      if (cnt == 0) first = n;
      ob[cnt++] = n;
    }
  }
  for (int j = cnt; j < ns; ++j) ob[j] = first;
}

// ---------------- fused SA stage: group -> MLP(3 layers, WMMA fp32) -> maxpool ----------------
// One block (4 wave32) per (b, s) group of 64 points. Each wave owns one 16-col N-tile.
// Layer-1 K is remapped to [xyz0, xyz1, xyz2, 0, feat0..] so each K-step has a uniform source.
// FMODE 0: feats = pointcloud[b][n][3+c] (row stride xstride). FMODE 1: feats = [b][c][Nsrc].
template <int CIN1, int CF, int FMODE, int COUT1, int COUT2, int COUT3>
__global__ __launch_bounds__(128) void sa_fused_kernel(
    const float* __restrict__ xyz_src, int xstride, int Nsrc,
    const float* __restrict__ feat_src,
    const float* __restrict__ new_xyz, const int* __restrict__ gidx, int S,
    const float* __restrict__ W1, const float* __restrict__ s1, const float* __restrict__ t1,
    const float* __restrict__ W2, const float* __restrict__ s2, const float* __restrict__ t2,
    const float* __restrict__ W3, const float* __restrict__ s3, const float* __restrict__ t3,
    float* __restrict__ out)   // (16, COUT3, S)
{
  __shared__ float bufA[128 * 64];   // layer-1 output (and pm scratch in layer 3)
  __shared__ float bufB[128 * 64];   // layer-2 output

  constexpr int CFP = (CF + 3) & ~3;   // padded feature-channel count
  constexpr int KP1 = 4 + CFP;         // remapped layer-1 K
  constexpr int MT1 = COUT1 / 16;
  constexpr int MT2 = COUT2 / 16;
  constexpr int MT3 = COUT3 / 16;

  const int bs = blockIdx.x;
  const int b = bs / S, s = bs % S;
  const int tid = threadIdx.x;
  const int wave = tid >> 5, lane = tid & 31;
  const int h = lane >> 4, ln = lane & 15;
  const int col = wave * 16 + ln;     // this lane's group column (B/C/D fragment N index)

  // per-lane gathered point and centered coords (layer-1 rows 0..2)
  const int pi = gidx[((size_t)b * S + s) * 64 + col];
  const float cx = new_xyz[((size_t)b * S + s) * 3 + 0];
  const float cy = new_xyz[((size_t)b * S + s) * 3 + 1];
  const float cz = new_xyz[((size_t)b * S + s) * 3 + 2];
  const float px = xyz_src[((size_t)b * Nsrc + pi) * xstride + 0] - cx;
  const float py = xyz_src[((size_t)b * Nsrc + pi) * xstride + 1] - cy;
  const float pz = xyz_src[((size_t)b * Nsrc + pi) * xstride + 2] - cz;

  auto featload = [&](int c) -> float {   // c already clamped to [0, CF)
    if (FMODE == 0) return feat_src[((size_t)b * Nsrc + pi) * xstride + 3 + c];
    else            return feat_src[((size_t)b * CF + c) * Nsrc + pi];
  };

  // ---- layer 1 : streamed X0, kb-outer / mt-inner, output -> bufA ----
  v8f acc1[MT1];
  #pragma unroll
  for (int mt = 0; mt < MT1; ++mt) acc1[mt] = v8f{0.f, 0.f, 0.f, 0.f, 0.f, 0.f, 0.f, 0.f};

  {  // kb = 0 : rows {px, py, pz, 0}; pure register selects, no loads for B
    v2f bf;
    bf.x = h ? pz : px;
    bf.y = h ? 0.f : py;
    #pragma unroll
    for (int mt = 0; mt < MT1; ++mt) {
      const float* wr = W1 + (size_t)(mt * 16 + ln) * CIN1;
      v2f a;
      a.x = wr[2 * h];                 // k'=0 -> col 0, k'=2 -> col 2
      a.y = h ? 0.f : wr[1];           // k'=3 -> zero pad column
      acc1[mt] = wmma4(a, bf, acc1[mt]);
    }
  }
  #pragma unroll
  for (int kb = 4; kb < KP1; kb += 4) {    // feature rows: c = kb-4 + 2h (+1)
    const int c0 = kb - 4 + 2 * h;
    const int c1 = c0 + 1;
    const bool ok0 = (c0 < CF), ok1 = (c1 < CF);
    const int cc0 = ok0 ? c0 : (CF - 1);   // clamp -> unconditional loads
    const int cc1 = ok1 ? c1 : (CF - 1);
    float f0 = featload(cc0);
    float f1 = featload(cc1);
    v2f bf;
    bf.x = ok0 ? f0 : 0.f;
    bf.y = ok1 ? f1 : 0.f;
    #pragma unroll
    for (int mt = 0; mt < MT1; ++mt) {
      const float* wr = W1 + (size_t)(mt * 16 + ln) * CIN1 + 3;  // feature columns of W1
      float w0 = wr[cc0];
      float w1 = wr[cc1];
      v2f a;
      a.x = ok0 ? w0 : 0.f;
      a.y = ok1 ? w1 : 0.f;
      acc1[mt] = wmma4(a, bf, acc1[mt]);
    }
  }
  #pragma unroll
  for (int mt = 0; mt < MT1; ++mt) {
    #pragma unroll
    for (int r = 0; r < 8; ++r) {
      int co = mt * 16 + r + 8 * h;
      bufA[co * 64 + col] = fmaxf(acc1[mt][r] * s1[co] + t1[co], 0.f);
    }
  }
  __syncthreads();

  // ---- layer 2 : bufA -> bufB ----
  #pragma unroll
  for (int mt = 0; mt < MT2; ++mt) {
    v8f acc = {0.f, 0.f, 0.f, 0.f, 0.f, 0.f, 0.f, 0.f};
    #pragma unroll 4
    for (int kb = 0; kb < COUT1; kb += 4) {
      const int k0 = kb + 2 * h;
      v2f a, bf;
      a.x = W2[(size_t)(mt * 16 + ln) * COUT1 + k0];
      a.y = W2[(size_t)(mt * 16 + ln) * COUT1 + k0 + 1];
      bf.x = bufA[k0 * 64 + col];
      bf.y = bufA[(k0 + 1) * 64 + col];
      acc = wmma4(a, bf, acc);
    }
    #pragma unroll
    for (int r = 0; r < 8; ++r) {
      int co = mt * 16 + r + 8 * h;
      bufB[co * 64 + col] = fmaxf(acc[r] * s2[co] + t2[co], 0.f);
    }
  }
  __syncthreads();

  // ---- layer 3 + streamed maxpool over the 64 points ----
  float* pm = bufA;                 // bufA free now; reuse 64 floats as cross-wave scratch
  #pragma unroll
  for (int mt = 0; mt < MT3; ++mt) {
    v8f acc = {0.f, 0.f, 0.f, 0.f, 0.f, 0.f, 0.f, 0.f};
    #pragma unroll 4
    for (int kb = 0; kb < COUT2; kb += 4) {
      const int k0 = kb + 2 * h;
      v2f a, bf;
      a.x = W3[(size_t)(mt * 16 + ln) * COUT2 + k0];
      a.y = W3[(size_t)(mt * 16 + ln) * COUT2 + k0 + 1];
      bf.x = bufB[k0 * 64 + col];
      bf.y = bufB[(k0 + 1) * 64 + col];
      acc = wmma4(a, bf, acc);
    }
    float v[8];
    #pragma unroll
    for (int r = 0; r < 8; ++r) {
      int co = mt * 16 + r + 8 * h;
      v[r] = fmaxf(acc[r] * s3[co] + t3[co], 0.f);
    }
    // max over the 16 columns held by this half-wave
    #pragma unroll
    for (int off = 1; off < 16; off <<= 1) {
      #pragma unroll
      for (int r = 0; r < 8; ++r) v[r] = fmaxf(v[r], __shfl_xor(v[r], off, 32));
    }
    if (ln == 0) {
      #pragma unroll
      for (int r = 0; r < 8; ++r) pm[wave * 16 + 8 * h + r] = v[r];
    }
    __syncthreads();
    if (tid < 16) {
      float m = fmaxf(fmaxf(pm[tid], pm[16 + tid]), fmaxf(pm[32 + tid], pm[48 + tid]));
      out[((size_t)b * COUT3 + mt * 16 + tid) * S + s] = m;
    }
    __syncthreads();
  }
}

// ---------------- generic one-wave WMMA GEMM tile: Y = act(A(MxK) @ X(b,Kx,P)) ----------------
__global__ __launch_bounds__(32) void gemm_tile_kernel(
    const float* __restrict__ A, const float* __restrict__ X, float* __restrict__ Y,
    const float* __restrict__ sArr, const float* __restrict__ tArr,
    int M, int K, int Kx, int P, int Mt, int Pt, int relu, int tflag)
{
  const int blk = blockIdx.x;
  const int ntiles = Mt * Pt;
  const int bb = blk / ntiles;
  const int r2 = blk % ntiles;
  const int mt = r2 / Pt, nt = r2 % Pt;
  const int lane = threadIdx.x;
  const int h = lane >> 4, ln = lane & 15;
  const float* Xb = X + (size_t)bb * Kx * P;
  const int pcol = nt * 16 + ln;
  const int arow = mt * 16 + ln;
  const int ar = (arow < M) ? arow : (M - 1);   // clamp -> branchless loads
  const bool rok = (arow < M);

  v8f acc = {0.f, 0.f, 0.f, 0.f, 0.f, 0.f, 0.f, 0.f};
  for (int kb = 0; kb < Kx; kb += 4) {
    const int k0 = kb + 2 * h;
    const bool k0ok = (k0 < K), k1ok = (k0 + 1 < K);
    const int kc0 = k0ok ? k0 : (K - 1);
    const int kc1 = k1ok ? (k0 + 1) : (K - 1);
    float a0 = A[(size_t)ar * K + kc0];
    float a1 = A[(size_t)ar * K + kc1];
    v2f a, bf;
    a.x = (rok && k0ok) ? a0 : 0.f;
    a.y = (rok && k1ok) ? a1 : 0.f;
    bf.x = Xb[(size_t)k0 * P + pcol];           // Kx rows exist; pad rows are zeroed
    bf.y = Xb[(size_t)(k0 + 1) * P + pcol];
    acc = wmma4(a, bf, acc);
  }
  #pragma unroll
  for (int r = 0; r < 8; ++r) {
    int co = mt * 16 + r + 8 * h;
    if (co < M) {
      float v = acc[r];
      if (sArr)      v = v * sArr[co] + tArr[co];
      else if (tArr) v = v + tArr[co];
      if (relu) v = fmaxf(v, 0.f);
      if (tflag) Y[(size_t)pcol * M + co] = v;                 // transposed store (P rows)
      else       Y[((size_t)bb * M + co) * P + pcol] = v;
    }
  }
}

// ---------------- SA3 input build: X3 (16, 260, 128), row 259 zero pad ----------------
__global__ void build_x3_kernel(const float* __restrict__ nx2, const float* __restrict__ sa2o,
                                float* __restrict__ X3) {
  int i = blockIdx.x * blockDim.x + threadIdx.x;
  const int TOT = 16 * 260 * 128;
  if (i >= TOT) return;
  int p = i % 128;
  int t = i / 128;
  int k = t % 260, b = t / 260;
  float v;
  if (k < 3)        v = nx2[((size_t)b * 128 + p) * 3 + k];
  else if (k < 259) v = sa2o[((size_t)b * 256 + (k - 3)) * 128 + p];
  else              v = 0.f;
  X3[((size_t)b * 260 + k) * 128 + p] = v;
}

// ---------------- global maxpool over 128 pts, store channel-major (1024,16) ----------------
__global__ void maxpool_kernel(const float* __restrict__ y, float* __restrict__ outT) {
  int i = blockIdx.x * blockDim.x + threadIdx.x;
  if (i >= 16 * 1024) return;
  int c = i % 1024, b = i / 1024;
  const float* row = y + ((size_t)b * 1024 + c) * 128;
  float m = row[0];
  for (int p = 1; p < 128; ++p) m = fmaxf(m, row[p]);
  outT[(size_t)c * 16 + b] = m;
}

extern "C" void kernel_launch(void* const* d_in, const int* in_sizes, int n_in,
                              void* d_out, int out_size, void* d_ws, size_t ws_size,
                              hipStream_t stream) {
  (void)in_sizes; (void)n_in; (void)out_size; (void)ws_size;
  const float* pc = (const float*)d_in[0];   // (16, 4096, 6)

  float* wsf = (float*)d_ws;
  size_t o = 0;
  float* sBN  = wsf + o; o += 3328;
  float* tBN  = wsf + o; o += 3328;
  float* nx1  = wsf + o; o += (size_t)16 * 512 * 3;
  float* nx2  = wsf + o; o += (size_t)16 * 128 * 3;
  float* sa1o = wsf + o; o += (size_t)16 * 128 * 512;
  float* sa2o = wsf + o; o += (size_t)16 * 256 * 128;
  float* X3   = wsf + o; o += (size_t)16 * 260 * 128;
  float* y3a  = wsf + o; o += (size_t)16 * 256 * 128;
  float* y3b  = wsf + o; o += (size_t)16 * 512 * 128;
  float* y3c  = wsf + o; o += (size_t)16 * 1024 * 128;
  float* plT  = wsf + o; o += (size_t)1024 * 16;
  float* f1T  = wsf + o; o += (size_t)512 * 16;
  float* f2T  = wsf + o; o += (size_t)256 * 16;
  int* wsi   = (int*)(wsf + o);
  int* fidx1 = wsi; wsi += 16 * 512;
  int* fidx2 = wsi; wsi += 16 * 128;
  int* ball1 = wsi; wsi += (size_t)16 * 512 * 64;
  int* ball2 = wsi; wsi += (size_t)16 * 128 * 64;

  // fold BN params (layer W index, channels, offset into sBN/tBN)
  const int bw[11]   = {1, 6, 11, 16, 21, 26, 31, 36, 41, 46, 51};
  const int bc[11]   = {64, 64, 128, 128, 128, 256, 256, 512, 1024, 512, 256};
  const int boff[11] = {0, 64, 128, 256, 384, 512, 768, 1024, 1536, 2560, 3072};
  for (int l = 0; l < 11; ++l) {
    bn_fold_kernel<<<(bc[l] + 63) / 64, 64, 0, stream>>>(
        (const float*)d_in[bw[l] + 1], (const float*)d_in[bw[l] + 2],
        (const float*)d_in[bw[l] + 3], (const float*)d_in[bw[l] + 4],
        sBN + boff[l], tBN + boff[l], bc[l]);
  }

  // ---- SA1: npoint=512, r=0.2, ns=64, MLP 6->64->64->128 ----
  fps_kernel<<<16, 256, 0, stream>>>(pc, 6, 4096, 512, fidx1);
  gather_xyz_kernel<<<(16 * 512 + 255) / 256, 256, 0, stream>>>(pc, 6, 4096, fidx1, 512, nx1);
  ball_query_kernel<<<(16 * 512 + 255) / 256, 256, 0, stream>>>(pc, 6, 4096, nx1, 512, 0.04f, 64, ball1);
  sa_fused_kernel<6, 3, 0, 64, 64, 128><<<16 * 512, 128, 0, stream>>>(
      pc, 6, 4096, pc, nx1, ball1, 512,
      (const float*)d_in[1],  sBN + 0,   tBN + 0,
      (const float*)d_in[6],  sBN + 64,  tBN + 64,
      (const float*)d_in[11], sBN + 128, tBN + 128,
      sa1o);

  // ---- SA2: npoint=128, r=0.4, ns=64, MLP 131->128->128->256 ----
  fps_kernel<<<16, 256, 0, stream>>>(nx1, 3, 512, 128, fidx2);
  gather_xyz_kernel<<<(16 * 128 + 255) / 256, 256, 0, stream>>>(nx1, 3, 512, fidx2, 128, nx2);
  ball_query_kernel<<<(16 * 128 + 255) / 256, 256, 0, stream>>>(nx1, 3, 512, nx2, 128, 0.16f, 64, ball2);
  sa_fused_kernel<131, 128, 1, 128, 128, 256><<<16 * 128, 128, 0, stream>>>(
      nx1, 3, 512, sa1o, nx2, ball2, 128,
      (const float*)d_in[16], sBN + 256, tBN + 256,
      (const float*)d_in[21], sBN + 384, tBN + 384,
      (const float*)d_in[26], sBN + 512, tBN + 512,
      sa2o);

  // ---- SA3 (global): MLP 259->256->512->1024, then maxpool over 128 pts ----
  build_x3_kernel<<<(16 * 260 * 128 + 255) / 256, 256, 0, stream>>>(nx2, sa2o, X3);
  gemm_tile_kernel<<<16 * 16 * 8, 32, 0, stream>>>((const float*)d_in[31], X3,  y3a, sBN + 768,  tBN + 768,  256, 259, 260, 128, 16, 8, 1, 0);
  gemm_tile_kernel<<<16 * 32 * 8, 32, 0, stream>>>((const float*)d_in[36], y3a, y3b, sBN + 1024, tBN + 1024, 512, 256, 256, 128, 32, 8, 1, 0);
  gemm_tile_kernel<<<16 * 64 * 8, 32, 0, stream>>>((const float*)d_in[41], y3b, y3c, sBN + 1536, tBN + 1536, 1024, 512, 512, 128, 64, 8, 1, 0);
  maxpool_kernel<<<(16 * 1024 + 255) / 256, 256, 0, stream>>>(y3c, plT);

  // ---- FC head: 1024->512->256->40 (P = 16 batch columns) ----
  gemm_tile_kernel<<<32, 32, 0, stream>>>((const float*)d_in[46], plT, f1T, sBN + 2560, tBN + 2560, 512, 1024, 1024, 16, 32, 1, 1, 0);
  gemm_tile_kernel<<<16, 32, 0, stream>>>((const float*)d_in[51], f1T, f2T, sBN + 3072, tBN + 3072, 256, 512,  512,  16, 16, 1, 1, 0);
  gemm_tile_kernel<<<3,  32, 0, stream>>>((const float*)d_in[56], f2T, (float*)d_out, nullptr, (const float*)d_in[57], 40, 256, 256, 16, 3, 1, 0, 1);
}